// SynGCN_65343632441911
// MI455X (gfx1250) — compile-verified
//
#include <hip/hip_runtime.h>
#include <math.h>

// ---------------- model dims (match reference) ----------------
#define B_   128
#define S_   128
#define HID  360      // H
#define H4   1440     // 4H
#define H2   720      // 2H
#define IN0  360      // DW+DP+DN
#define DD_  64
#define AA_  200
#define CC_  42
#define NTOK (B_*S_)  // 16384

typedef float v2f __attribute__((ext_vector_type(2)));
typedef float v8f __attribute__((ext_vector_type(8)));

__device__ __forceinline__ float sigm_(float x) { return 1.0f / (1.0f + __expf(-x)); }

// ---------------- CDNA5 async global->LDS copy (guarded) ----------------
#if defined(__has_builtin)
#if __has_builtin(__builtin_amdgcn_global_load_async_to_lds_b128) && \
    __has_builtin(__builtin_amdgcn_s_wait_asynccnt)
#define ASYNC_LDS 1
#endif
#endif

#ifdef ASYNC_LDS
// builtin signature (from hipcc diagnostic): param0 = int4 addrspace(1)*,
// param1 = int4 addrspace(3)*, then imm offset, imm cpol.
typedef int v4i_ __attribute__((vector_size(16)));
typedef __attribute__((address_space(1))) v4i_ v4i_as1;
typedef __attribute__((address_space(3))) v4i_ v4i_as3;
// copy 16 bytes global -> LDS, tracked by ASYNCcnt
__device__ __forceinline__ void async_cp16(const float* gsrc, float* ldst) {
  __builtin_amdgcn_global_load_async_to_lds_b128(
      (v4i_as1*)(uintptr_t)gsrc,
      (v4i_as3*)(unsigned int)(uintptr_t)ldst, 0, 0);
}
__device__ __forceinline__ void wait_async0() {
  __builtin_amdgcn_s_wait_asynccnt(0);
}
#else
__device__ __forceinline__ void async_cp16(const float* gsrc, float* ldst) {
  *(float4*)ldst = *(const float4*)gsrc;
}
__device__ __forceinline__ void wait_async0() {}
#endif

// =====================================================================
// Generic WMMA GEMM:  C[M,N] = A[M,K] * W[N,K]^T  (+ bias[N])
// fp32 end-to-end via V_WMMA_F32_16X16X4_F32.
// Requirements: M % 64 == 0, K % 4 == 0. N edge-masked.
// Block = 256 threads = 8 waves -> 4 M-tiles x 2 N-groups; each wave owns
// TWO adjacent 16x16 N-tiles sharing one A fragment (3 b64 loads / 2 wmma).
// Operand layout per CDNA5 ISA 7.12.2:
//   A (16x4): lane l<16 holds M=l, K={0,1}; lane>=16 holds M=l-16, K={2,3}
//   B (4x16): lanes 0-15 -> K=v, N=lane; lanes 16-31 -> K=v+2
//   D (16x16): vgpr r, lanes 0-15 -> M=r, lanes 16-31 -> M=r+8; N=lane&15
// =====================================================================
__global__ __launch_bounds__(256) void gemm_wmma_nt(
    const float* __restrict__ A, int lda,
    const float* __restrict__ W, int ldw,
    const float* __restrict__ bias,
    float* __restrict__ C, int ldc,
    int M, int N, int K)
{
  const int lane = threadIdx.x & 31;
  const int wave = threadIdx.x >> 5;
  const int mBase = blockIdx.y * 64 + (wave & 3) * 16;
  const int nBase = blockIdx.x * 64 + (wave >> 2) * 32;   // 2 tiles / wave
  if (mBase >= M || nBase >= N) return;    // wave-uniform early out

  const int l15  = lane & 15;
  const int koff = (lane >> 4) << 1;       // 0 or 2
  const int n0 = nBase + l15;
  const int n1 = nBase + 16 + l15;
  const bool ok0 = (n0 < N), ok1 = (n1 < N);
  const bool fullTile = (nBase + 31 < N);  // wave-uniform

  const float* __restrict__ arow  = A + (size_t)(mBase + l15) * lda;
  const float* __restrict__ wrow0 = W + (size_t)(ok0 ? n0 : 0) * ldw;
  const float* __restrict__ wrow1 = W + (size_t)(ok1 ? n1 : 0) * ldw;

  v8f acc0 = {0.f,0.f,0.f,0.f,0.f,0.f,0.f,0.f};
  v8f acc1 = {0.f,0.f,0.f,0.f,0.f,0.f,0.f,0.f};

  if (fullTile) {
    for (int k = 0; k < K; k += 4) {
      v2f a, b0, b1;
      a.x  = arow[k + koff];  a.y  = arow[k + koff + 1];
      b0.x = wrow0[k + koff]; b0.y = wrow0[k + koff + 1];
      b1.x = wrow1[k + koff]; b1.y = wrow1[k + koff + 1];
      acc0 = __builtin_amdgcn_wmma_f32_16x16x4_f32(false, a, false, b0, (short)0, acc0, false, false);
      acc1 = __builtin_amdgcn_wmma_f32_16x16x4_f32(false, a, false, b1, (short)0, acc1, false, false);
    }
  } else {
    for (int k = 0; k < K; k += 4) {
      v2f a, b0, b1;
      a.x = arow[k + koff]; a.y = arow[k + koff + 1];
      float b0x = wrow0[k + koff], b0y = wrow0[k + koff + 1];
      float b1x = wrow1[k + koff], b1y = wrow1[k + koff + 1];
      if (!ok0) { b0x = 0.f; b0y = 0.f; }
      if (!ok1) { b1x = 0.f; b1y = 0.f; }
      b0.x = b0x; b0.y = b0y; b1.x = b1x; b1.y = b1y;
      acc0 = __builtin_amdgcn_wmma_f32_16x16x4_f32(false, a, false, b0, (short)0, acc0, false, false);
      acc1 = __builtin_amdgcn_wmma_f32_16x16x4_f32(false, a, false, b1, (short)0, acc1, false, false);
    }
  }

  const int rowAdd = (lane >> 4) << 3;     // 0 or 8
  if (ok0) {
    const float badd = bias ? bias[n0] : 0.f;
#pragma unroll
    for (int r = 0; r < 8; ++r)
      C[(size_t)(mBase + r + rowAdd) * ldc + n0] = acc0[r] + badd;
  }
  if (ok1) {
    const float badd = bias ? bias[n1] : 0.f;
#pragma unroll
    for (int r = 0; r < 8; ++r)
      C[(size_t)(mBase + r + rowAdd) * ldc + n1] = acc1[r] + badd;
  }
}

// =====================================================================
// One LSTM timestep, both directions fused.
// grid = 16 blocks: blocks 0..7 = forward batch-tiles, 8..15 = backward.
// Each block owns 16 batch rows:
//   phase 0: async-stage h_prev tile (16xHID, 23KB) into LDS
//   phase 1: WMMA GEMM h_prev*Whh^T (+xs[t]) -> LDS gate buffer (16xH4, 90KB)
//            (waves own PAIRS of N-tiles sharing the LDS A fragment)
//   phase 2: fused sigmoid/tanh gate math, c/h update, write [B,S,2H] output
// =====================================================================
__global__ __launch_bounds__(256) void lstm_step(
    const float* __restrict__ xs_f, const float* __restrict__ xs_b, // [B,S,H4]
    const float* __restrict__ whh_f, const float* __restrict__ whh_b, // [H4,HID]
    float* __restrict__ hs_f, float* __restrict__ cs_f,   // [B,HID]
    float* __restrict__ hs_b, float* __restrict__ cs_b,
    float* __restrict__ hout,                             // [B,S,H2]
    int t)
{
  extern __shared__ float smem[];
  float* hA = smem;              // [16][HID]  staged h_prev
  float* g  = smem + 16 * HID;   // [16][H4]   gate pre-activations

  const int lane = threadIdx.x & 31;
  const int wave = threadIdx.x >> 5;
  const int dir  = blockIdx.x >> 3;
  const int mBase = (blockIdx.x & 7) * 16;

  const float* __restrict__ xs  = dir ? xs_b  : xs_f;
  const float* __restrict__ whh = dir ? whh_b : whh_f;
  float* __restrict__ hs = dir ? hs_b : hs_f;
  float* __restrict__ cs = dir ? cs_b : cs_f;
  const int tt = dir ? (S_ - 1 - t) : t;

  // ---- phase 0: stage h_prev rows (contiguous 16*HID block) into LDS ----
  {
    const float* hsrc = hs + (size_t)mBase * HID;
    for (int i = threadIdx.x * 4; i < 16 * HID; i += 256 * 4)
      async_cp16(hsrc + i, hA + i);
    wait_async0();
  }
  __syncthreads();

  const int l15  = lane & 15;
  const int koff = (lane >> 4) << 1;
  const int rowAdd = (lane >> 4) << 3;
  const float* arowL = hA + l15 * HID;   // LDS reads -> ds_load_b64

  // ---- phase 1: gates = h_prev @ Whh^T + xs[:,tt,:] ----
  for (int p = wave; p < (H4 / 16) / 2; p += 8) {   // 45 tile-pairs
    const int nBase = p * 32;
    const float* __restrict__ wrow0 = whh + (size_t)(nBase + l15) * HID;
    const float* __restrict__ wrow1 = whh + (size_t)(nBase + 16 + l15) * HID;
    v8f acc0 = {0.f,0.f,0.f,0.f,0.f,0.f,0.f,0.f};
    v8f acc1 = {0.f,0.f,0.f,0.f,0.f,0.f,0.f,0.f};
    for (int k = 0; k < HID; k += 4) {
      v2f a, b0, b1;
      a.x  = arowL[k + koff];  a.y  = arowL[k + koff + 1];
      b0.x = wrow0[k + koff];  b0.y = wrow0[k + koff + 1];
      b1.x = wrow1[k + koff];  b1.y = wrow1[k + koff + 1];
      acc0 = __builtin_amdgcn_wmma_f32_16x16x4_f32(false, a, false, b0, (short)0, acc0, false, false);
      acc1 = __builtin_amdgcn_wmma_f32_16x16x4_f32(false, a, false, b1, (short)0, acc1, false, false);
    }
    const int n0 = nBase + l15, n1 = nBase + 16 + l15;
#pragma unroll
    for (int r = 0; r < 8; ++r) {
      const int lb = r + rowAdd;                 // local batch row 0..15
      const size_t xrow = ((size_t)(mBase + lb) * S_ + tt) * H4;
      g[lb * H4 + n0] = acc0[r] + xs[xrow + n0];
      g[lb * H4 + n1] = acc1[r] + xs[xrow + n1];
    }
  }
  __syncthreads();

  // ---- phase 2: i,f,g,o -> c,h ----
  for (int idx = threadIdx.x; idx < 16 * HID; idx += 256) {
    const int lb = idx / HID;
    const int hh = idx - lb * HID;
    const int bG = mBase + lb;
    const float gi = g[lb * H4 + hh];
    const float gf = g[lb * H4 + HID   + hh];
    const float gg = g[lb * H4 + 2*HID + hh];
    const float go = g[lb * H4 + 3*HID + hh];
    const size_t si = (size_t)bG * HID + hh;
    const float c = sigm_(gf) * cs[si] + sigm_(gi) * tanhf(gg);
    const float h = sigm_(go) * tanhf(c);
    cs[si] = c;
    hs[si] = h;
    hout[((size_t)bG * S_ + tt) * H2 + dir * HID + hh] = h;
  }
}

// ---------------- small helper kernels ----------------
__global__ void zero_f32(float* p, size_t n) {
  size_t i = (size_t)blockIdx.x * blockDim.x + threadIdx.x;
  if (i < n) p[i] = 0.f;
}

// token-wise embedding gather: x[tok] = [emb|pos|ner], dp[tok] = deprel_emb
__global__ __launch_bounds__(128) void gather_embed(
    const int* __restrict__ words, const int* __restrict__ pos,
    const int* __restrict__ ner,   const int* __restrict__ dep,
    const float* __restrict__ emb,  const float* __restrict__ pemb,
    const float* __restrict__ nemb, const float* __restrict__ demb,
    float* __restrict__ x, float* __restrict__ dp)
{
  const int tok = blockIdx.x;
  const int tid = threadIdx.x;
  const float* e = emb + (size_t)words[tok] * 300;
  float* xr = x + (size_t)tok * IN0;
  for (int i = tid; i < 300; i += 128) xr[i] = e[i];
  const float* p = pemb + (size_t)pos[tok] * 30;
  for (int i = tid; i < 30; i += 128) xr[300 + i] = p[i];
  const float* nn = nemb + (size_t)ner[tok] * 30;
  for (int i = tid; i < 30; i += 128) xr[330 + i] = nn[i];
  const float* d = demb + (size_t)dep[tok] * DD_;
  float* dr = dp + (size_t)tok * DD_;
  for (int i = tid; i < DD_; i += 128) dr[i] = d[i];
}

// masked mean pool over sequence: dst[b, dstOff + h] = mean_{s: !mask} feats[b,s,h]
__global__ __launch_bounds__(256) void pool_span(
    const float* __restrict__ feats,        // [B,S,H2]
    const unsigned char* __restrict__ mask, // [B,S]
    float* __restrict__ dst, int dstOff)    // [B,H4]
{
  __shared__ float red[256];
  const int b = blockIdx.x, tid = threadIdx.x;
  float c = 0.f;
  if (tid < S_) c = mask[b * S_ + tid] ? 0.f : 1.f;
  red[tid] = c; __syncthreads();
  for (int o = 128; o; o >>= 1) { if (tid < o) red[tid] += red[tid + o]; __syncthreads(); }
  const float inv = 1.0f / red[0];
  for (int h = tid; h < H2; h += 256) {
    float acc = 0.f;
    for (int s = 0; s < S_; ++s)
      if (!mask[b * S_ + s]) acc += feats[((size_t)b * S_ + s) * H2 + h];
    dst[(size_t)b * H4 + dstOff + h] = acc * inv;
  }
}

// scores[b,s] = sum_a tanh(G1[tok,a] + qv[b,a]) * tw[a] + tb ; masked -> -inf
__global__ __launch_bounds__(256) void attn_scores(
    const float* __restrict__ G1, const float* __restrict__ qv,
    const float* __restrict__ tw, const float* __restrict__ tb,
    const unsigned char* __restrict__ dmask, float* __restrict__ scores)
{
  const int lane = threadIdx.x & 31;
  const int wave = threadIdx.x >> 5;
  const int idx = blockIdx.x * 8 + wave;   // token id (b*S+s)
  const int b = idx >> 7;
  float part = 0.f;
  for (int a = lane; a < AA_; a += 32)
    part += tanhf(G1[(size_t)idx * AA_ + a] + qv[(size_t)b * AA_ + a]) * tw[a];
  for (int off = 16; off; off >>= 1) part += __shfl_down(part, off);
  if (lane == 0) {
    float sc = part + tb[0];
    if (dmask[idx]) sc = -INFINITY;
    scores[idx] = sc;
  }
}

__global__ __launch_bounds__(128) void softmax_s(
    const float* __restrict__ scores, float* __restrict__ w)
{
  __shared__ float red[128];
  const int b = blockIdx.x, s = threadIdx.x;
  const float v = scores[b * S_ + s];
  red[s] = v; __syncthreads();
  for (int o = 64; o; o >>= 1) { if (s < o) red[s] = fmaxf(red[s], red[s + o]); __syncthreads(); }
  const float m = red[0]; __syncthreads();
  const float e = __expf(v - m);
  red[s] = e; __syncthreads();
  for (int o = 64; o; o >>= 1) { if (s < o) red[s] += red[s + o]; __syncthreads(); }
  w[b * S_ + s] = e / red[0];
}

// ---------------- GCN ----------------
__global__ void deg_init(float* __restrict__ deg) {
  const int i = blockIdx.x * 256 + threadIdx.x;
  if (i < NTOK) deg[i] = 1.0f;               // self loop weight
}
__global__ void deg_scatter(const long long* __restrict__ ei,
                            const float* __restrict__ w, float* __restrict__ deg) {
  const int e = blockIdx.x * 256 + threadIdx.x;
  if (e < NTOK) atomicAdd(&deg[ei[NTOK + e]], w[e]);   // dst row of edge_index
}
__global__ void dinv_k(const float* __restrict__ deg, float* __restrict__ dinv) {
  const int i = blockIdx.x * 256 + threadIdx.x;
  if (i < NTOK) dinv[i] = deg[i] > 0.f ? rsqrtf(deg[i]) : 0.f;
}
// self-loop term + bias:  gout[n,h] = dinv[n]^2 * xl[n,h] + b[h]
__global__ void gcn_self(const float* __restrict__ xl, const float* __restrict__ dinv,
                         const float* __restrict__ bias, float* __restrict__ gout) {
  const size_t i = (size_t)blockIdx.x * 256 + threadIdx.x;
  if (i < (size_t)NTOK * H2) {
    const int n = (int)(i / H2);
    const int h = (int)(i - (size_t)n * H2);
    gout[i] = dinv[n] * dinv[n] * xl[i] + bias[h];
  }
}
// edge scatter:  gout[dst] += dinv[src]*w[e]*dinv[dst] * xl[src]
__global__ __launch_bounds__(256) void gcn_edge(
    const long long* __restrict__ ei, const float* __restrict__ w,
    const float* __restrict__ dinv, const float* __restrict__ xl,
    float* __restrict__ gout)
{
  const int e = blockIdx.x;
  const long long src = ei[e];
  const long long dst = ei[NTOK + e];
  const float nrm = dinv[src] * w[e] * dinv[dst];
  const float* __restrict__ xsr = xl + (size_t)src * H2;
  float* __restrict__ od = gout + (size_t)dst * H2;
  for (int h = threadIdx.x; h < H2; h += 256)
    atomicAdd(&od[h], nrm * xsr[h]);
}

// =====================================================================
// host orchestration
// =====================================================================
extern "C" void kernel_launch(void* const* d_in, const int* in_sizes, int n_in,
                              void* d_out, int out_size, void* d_ws, size_t ws_size,
                              hipStream_t stream) {
  (void)in_sizes; (void)n_in; (void)out_size; (void)ws_size;

  // ----- inputs (setup_inputs dict order) -----
  const int*           words   = (const int*)d_in[0];
  const int*           pos     = (const int*)d_in[1];
  const int*           ner     = (const int*)d_in[2];
  const int*           deprel  = (const int*)d_in[3];
  const unsigned char* dmask   = (const unsigned char*)d_in[4];
  const unsigned char* smask   = (const unsigned char*)d_in[5];
  const unsigned char* omask   = (const unsigned char*)d_in[6];
  const long long*     eidx    = (const long long*)d_in[7];
  const float *emb = (const float*)d_in[8],  *pemb = (const float*)d_in[9];
  const float *nemb = (const float*)d_in[10], *demb = (const float*)d_in[11];
  const float *wih0f = (const float*)d_in[12], *whh0f = (const float*)d_in[13], *b0f = (const float*)d_in[14];
  const float *wih0b = (const float*)d_in[15], *whh0b = (const float*)d_in[16], *b0b = (const float*)d_in[17];
  const float *wih1f = (const float*)d_in[18], *whh1f = (const float*)d_in[19], *b1f = (const float*)d_in[20];
  const float *wih1b = (const float*)d_in[21], *whh1b = (const float*)d_in[22], *b1b = (const float*)d_in[23];
  const float *auw = (const float*)d_in[24], *aub = (const float*)d_in[25];
  const float *avw = (const float*)d_in[26], *atw = (const float*)d_in[27], *atb = (const float*)d_in[28];
  const float *gcnw = (const float*)d_in[29], *gcnb = (const float*)d_in[30];
  const float *linw = (const float*)d_in[31], *linb = (const float*)d_in[32];
  float* out = (float*)d_out;

  // ----- workspace carve-up (floats) -----
  float* ws = (float*)d_ws;
  size_t off = 0;
  auto alloc = [&](size_t n) { float* p = ws + off; off += n; return p; };
  float* x    = alloc((size_t)NTOK * IN0);   // embedded input
  float* dp   = alloc((size_t)NTOK * DD_);   // deprel embeddings
  float* xsf  = alloc((size_t)NTOK * H4);    // input-proj fwd (reused, then as xl)
  float* xsb  = alloc((size_t)NTOK * H4);    // input-proj bwd (reused, then as gout)
  float* h0   = alloc((size_t)NTOK * H2);    // layer-0 output
  float* outb = alloc((size_t)NTOK * H2);    // layer-1 output (GCN input nodes)
  float* hsf  = alloc((size_t)B_ * HID);     // 4 consecutive state buffers
  float* csf  = alloc((size_t)B_ * HID);
  float* hsb  = alloc((size_t)B_ * HID);
  float* csb  = alloc((size_t)B_ * HID);
  float* q    = alloc((size_t)B_ * H4);      // [subj|obj]
  float* G1   = alloc((size_t)NTOK * AA_);
  float* qv   = alloc((size_t)B_ * AA_);
  float* scor = alloc((size_t)NTOK);
  float* wts  = alloc((size_t)NTOK);
  float* deg  = alloc((size_t)NTOK);
  float* dnv  = alloc((size_t)NTOK);
  float* fh   = alloc((size_t)B_ * H4);
  float* xl   = xsf;    // reuse: input-proj buffers dead after layer-1 recurrence
  float* gout = xsb;

  const size_t LSTM_LDS = (16 * HID + 16 * H4) * sizeof(float);  // 115200 B
  (void)hipFuncSetAttribute((const void*)lstm_step,
                            hipFuncAttributeMaxDynamicSharedMemorySize, (int)LSTM_LDS);

  auto gemm = [&](const float* A, int lda, const float* W, int ldw, const float* bias,
                  float* C, int ldc, int M, int N, int K) {
    dim3 g((N + 63) / 64, M / 64);
    gemm_wmma_nt<<<g, 256, 0, stream>>>(A, lda, W, ldw, bias, C, ldc, M, N, K);
  };

  // 1) embedding gathers
  gather_embed<<<NTOK, 128, 0, stream>>>(words, pos, ner, deprel,
                                         emb, pemb, nemb, demb, x, dp);

  // 2) layer-0 input projections: xs = x @ Wih^T + b   [NTOK,H4]
  gemm(x, IN0, wih0f, IN0, b0f, xsf, H4, NTOK, H4, IN0);
  gemm(x, IN0, wih0b, IN0, b0b, xsb, H4, NTOK, H4, IN0);

  // 3) layer-0 recurrence (h/c states zeroed; states are contiguous)
  zero_f32<<<(4 * B_ * HID + 255) / 256, 256, 0, stream>>>(hsf, (size_t)4 * B_ * HID);
  for (int t = 0; t < S_; ++t)
    lstm_step<<<16, 256, LSTM_LDS, stream>>>(xsf, xsb, whh0f, whh0b,
                                             hsf, csf, hsb, csb, h0, t);

  // 4) layer-1 input projections (overwrite xs buffers)
  gemm(h0, H2, wih1f, H2, b1f, xsf, H4, NTOK, H4, H2);
  gemm(h0, H2, wih1b, H2, b1b, xsb, H4, NTOK, H4, H2);

  // 5) layer-1 recurrence
  zero_f32<<<(4 * B_ * HID + 255) / 256, 256, 0, stream>>>(hsf, (size_t)4 * B_ * HID);
  for (int t = 0; t < S_; ++t)
    lstm_step<<<16, 256, LSTM_LDS, stream>>>(xsf, xsb, whh1f, whh1b,
                                             hsf, csf, hsb, csb, outb, t);

  // 6) subj/obj pooling -> q = [subj|obj]  [B,H4]
  pool_span<<<B_, 256, 0, stream>>>(outb, smask, q, 0);
  pool_span<<<B_, 256, 0, stream>>>(outb, omask, q, H2);

  // 7) attention: G1 = dp @ Uw^T + ub ; qv = q @ Vw^T
  gemm(dp, DD_, auw, DD_, aub, G1, AA_, NTOK, AA_, DD_);
  gemm(q, H4, avw, H4, nullptr, qv, AA_, B_, AA_, H4);
  attn_scores<<<NTOK / 8, 256, 0, stream>>>(G1, qv, atw, atb, dmask, scor);
  softmax_s<<<B_, 128, 0, stream>>>(scor, wts);

  // 8) GCN: xl = nodes @ gcn_w^T ; symmetric-norm aggregation with self loops
  gemm(outb, H2, gcnw, H2, nullptr, xl, H2, NTOK, H2, H2);
  deg_init<<<(NTOK + 255) / 256, 256, 0, stream>>>(deg);
  deg_scatter<<<(NTOK + 255) / 256, 256, 0, stream>>>(eidx, wts, deg);
  dinv_k<<<(NTOK + 255) / 256, 256, 0, stream>>>(deg, dnv);
  gcn_self<<<(int)(((size_t)NTOK * H2 + 255) / 256), 256, 0, stream>>>(xl, dnv, gcnb, gout);
  gcn_edge<<<NTOK, 256, 0, stream>>>(eidx, wts, dnv, xl, gout);

  // 9) final pooling + classifier: logits = [subj2|obj2] @ lin_w^T + lin_b
  pool_span<<<B_, 256, 0, stream>>>(gout, smask, fh, 0);
  pool_span<<<B_, 256, 0, stream>>>(gout, omask, fh, H2);
  gemm(fh, H4, linw, H4, linb, out, CC_, B_, CC_, H4);
}